// ElementsFeatureProcessor_70798240907696
// MI455X (gfx1250) — compile-verified
//
#include <hip/hip_runtime.h>

typedef __attribute__((ext_vector_type(2))) float v2f;
typedef __attribute__((ext_vector_type(4))) float v4f;
typedef __attribute__((ext_vector_type(8))) float v8f;

#define RARE_MIN 57
#define RARE_MAX 80

// One wave handles 32 tokens per group:
//   - two 16-token WMMA tiles compute relu(W @ feats + b) for 16 features
//   - every lane gathers one token's 8-float embedding
// D = A(16x4 f32) x B(4x16 f32) + C via V_WMMA_F32_16X16X4_F32, K padded 5->8.
__global__ __launch_bounds__(256) void elements_feature_kernel(
    const float* __restrict__ info,   // (T, 7): 5 feats, atomic_num, elem_type
    const float* __restrict__ mask,   // (T,)
    const float* __restrict__ W,      // (16, 5) row-major
    const float* __restrict__ bvec,   // (16,)
    const float* __restrict__ emb,    // (25, 8)
    float* __restrict__ out,          // (T, 24)
    int num_groups)                   // T / 32
{
  const int lane = threadIdx.x & 31;
  const int l    = lane & 15;        // M index (A) / token-in-tile (B, D)
  const int half = lane >> 4;        // K half for A/B; feature half for C/D
  const int wave   = blockIdx.x * (blockDim.x >> 5) + (threadIdx.x >> 5);
  const int nwaves = gridDim.x * (blockDim.x >> 5);

  // --- A operands (weights), loop-invariant per wave ---
  // 32-bit A 16x4 layout: lanes 0-15 -> (K0,K1), lanes 16-31 -> (K2,K3).
  v2f a1, a2;
  {
    const float* wr = W + l * 5;
    if (half == 0) { a1.x = wr[0]; a1.y = wr[1]; a2.x = wr[4]; a2.y = 0.0f; }
    else           { a1.x = wr[2]; a1.y = wr[3]; a2.x = 0.0f;  a2.y = 0.0f; }
  }
  // C/D layout: VGPR j holds feature j (lanes 0-15) / j+8 (lanes 16-31).
  float bias[8];
  #pragma unroll
  for (int j = 0; j < 8; ++j) bias[j] = bvec[j + 8 * half];

  for (int g = wave; g < num_groups; g += nwaves) {
    const int base = g * 32;

    #pragma unroll
    for (int tile = 0; tile < 2; ++tile) {
      const int tok = base + tile * 16 + l;
      const float m = mask[tok];
      const float* fi = info + (long)tok * 7;

      // B 4x16 layout mirrors A: lanes 0-15 -> (K0,K1), 16-31 -> (K2,K3).
      v2f b1, b2;
      if (half == 0) { b1.x = fi[0] * m; b1.y = fi[1] * m; b2.x = fi[4] * m; b2.y = 0.0f; }
      else           { b1.x = fi[2] * m; b1.y = fi[3] * m; b2.x = 0.0f;      b2.y = 0.0f; }

      v8f acc;
      #pragma unroll
      for (int j = 0; j < 8; ++j) acc[j] = bias[j];

      // 8 args: (neg_a, A, neg_b, B, c_mod, C, reuse_a, reuse_b)
      acc = __builtin_amdgcn_wmma_f32_16x16x4_f32(
          false, a1, false, b1, (short)0, acc, false, false);
      acc = __builtin_amdgcn_wmma_f32_16x16x4_f32(
          false, a2, false, b2, (short)0, acc, false, false);

      // relu(.)*mask; this lane's D entries all belong to token `tok`.
      v4f s0, s1;
      #pragma unroll
      for (int j = 0; j < 4; ++j) {
        s0[j] = acc[j]     > 0.0f ? acc[j]     * m : 0.0f;
        s1[j] = acc[j + 4] > 0.0f ? acc[j + 4] * m : 0.0f;
      }
      float* op = out + (long)tok * 24 + 8 * half;   // 16B-aligned
      *(v4f*)(op)     = s0;
      *(v4f*)(op + 4) = s1;
    }

    // --- embedding gather: one token per lane ---
    {
      const int t = base + lane;
      const float m  = mask[t];
      const float zf = info[(long)t * 7 + 5] * m;
      const int zi   = (int)zf;
      const int idx  = (zi >= RARE_MIN && zi <= RARE_MAX) ? (zi - RARE_MIN + 1) : 0;
      const float* ep = emb + idx * 8;               // 32B stride -> 16B aligned
      v4f e0 = *(const v4f*)(ep);
      v4f e1 = *(const v4f*)(ep + 4);
      #pragma unroll
      for (int j = 0; j < 4; ++j) { e0[j] *= m; e1[j] *= m; }
      float* op = out + (long)t * 24 + 16;           // 64B offset -> aligned
      *(v4f*)(op)     = e0;
      *(v4f*)(op + 4) = e1;
    }
  }
}

extern "C" void kernel_launch(void* const* d_in, const int* in_sizes, int n_in,
                              void* d_out, int out_size, void* d_ws, size_t ws_size,
                              hipStream_t stream) {
  const float* info = (const float*)d_in[0];  // (B*N*7,)
  const float* mask = (const float*)d_in[1];  // (B*N,)
  const float* W    = (const float*)d_in[2];  // (80,)
  const float* b    = (const float*)d_in[3];  // (16,)
  const float* emb  = (const float*)d_in[4];  // (200,)
  float* out = (float*)d_out;                 // (B*N*24,)

  const int T = in_sizes[1];                  // B*N tokens (2,097,152)
  const int num_groups = T / 32;              // 65,536 (T divisible by 32)

  const int threads = 256;                    // 8 waves/block
  int blocks = (num_groups + 7) / 8;          // one 32-token group per wave
  if (blocks > 8192) blocks = 8192;           // grid-stride covers the rest
  if (blocks < 1) blocks = 1;

  elements_feature_kernel<<<blocks, threads, 0, stream>>>(
      info, mask, W, b, emb, out, num_groups);
}